// SemiGNN_65730179498118
// MI455X (gfx1250) — compile-verified
//
#include <hip/hip_runtime.h>
#include <cstdint>
#include <cstddef>

#define Nn   50000
#define Vv   3
#define DIMd 128
#define Hh   64
#define VH   (Vv * Hh)          /* 192 */

typedef __attribute__((ext_vector_type(2)))  float    v2f;
typedef __attribute__((ext_vector_type(8)))  float    v8f;

// ---- ordered-uint encoding for float atomicMax ----
__device__ __forceinline__ unsigned enc_f(float f) {
    unsigned u = __float_as_uint(f);
    return (u & 0x80000000u) ? ~u : (u | 0x80000000u);
}
__device__ __forceinline__ float dec_f(unsigned u) {
    return (u & 0x80000000u) ? __uint_as_float(u & 0x7fffffffu) : __uint_as_float(~u);
}

__device__ __forceinline__ float attn_nv(const float* __restrict__ scores,
                                         const unsigned* __restrict__ gmax,
                                         const float* __restrict__ gsum,
                                         int n, int v) {
    float m = dec_f(gmax[v]);
    return expf(scores[n * Vv + v] - m) / gsum[v];
}

// ---------------- K0: init workspace ----------------
__global__ __launch_bounds__(256) void k_init(float* agg, float* deg, unsigned* gmax,
                                              float* gsum, float* avg) {
    size_t i = (size_t)blockIdx.x * blockDim.x + threadIdx.x;
    if (i < (size_t)Nn * VH) agg[i] = 0.0f;
    if (i < Nn)  deg[i] = 1.0f;        // self-loop counted in degree
    if (i < Vv)  { gmax[i] = 0u; gsum[i] = 0.0f; }
    if (i < VH)  avg[i] = 0.0f;
}

// ---------------- K1: h = einsum('nvd,vdh->nvh') via f32 WMMA ----------------
// One block = one view x 8 node-tiles (8 waves). Wv[v] (32 KB) is staged into
// LDS once per block with gfx1250 async global->LDS B128 copies, then each
// wave computes a 16(node) x 64(H) tile with V_WMMA_F32_16X16X4_F32 reading
// B-fragments from LDS.
__global__ __launch_bounds__(256) void k_gemm_wmma(const float* __restrict__ x,
                                                   const float* __restrict__ Wv,
                                                   float* __restrict__ hbuf) {
    __shared__ float sw[DIMd * Hh];                 // 32 KB staged weights
    const int tilesPerView  = Nn / 16;              // 3125
    const int blocksPerView = (tilesPerView + 7) / 8; // 391
    const int v   = blockIdx.x / blocksPerView;
    const int tb  = blockIdx.x % blocksPerView;
    const int tid = threadIdx.x;

    const float* wbase = Wv + (size_t)v * DIMd * Hh;

    // ---- async stage Wv[v] -> LDS (2048 x 16B chunks, 8 per thread) ----
    {
        unsigned lds_base = (unsigned)(uintptr_t)(void*)sw;   // low 32b = LDS offset
        #pragma unroll
        for (int i = 0; i < (DIMd * Hh) / (256 * 4); ++i) {   // 8 iterations
            int chunk = tid + i * 256;                        // 16-byte chunk id
            unsigned laddr = lds_base + (unsigned)chunk * 16u;
            const float* gp = wbase + (size_t)chunk * 4;
            asm volatile("global_load_async_to_lds_b128 %0, %1, off"
                         :: "v"(laddr), "v"(gp) : "memory");
        }
        asm volatile("s_wait_asynccnt 0x0" ::: "memory");
    }
    __syncthreads();

    const int wave = tid >> 5;
    const int lane = tid & 31;
    const int tile = tb * 8 + wave;
    if (tile >= tilesPerView) return;               // uniform per wave, after barrier

    const int node0 = tile * 16;
    const int g     = lane >> 4;                    // lane group 0/1
    const int r     = lane & 15;                    // row/col within group

    const float* xrow = x + (size_t)(node0 + r) * (Vv * DIMd) + (size_t)v * DIMd;

    v8f acc[4];
    #pragma unroll
    for (int t = 0; t < 4; ++t)
        #pragma unroll
        for (int j = 0; j < 8; ++j) acc[t][j] = 0.0f;

    // V_WMMA_F32_16X16X4_F32: A 16x4 f32, VGPR j holds K = k + j + 2*g.
    for (int k = 0; k < DIMd; k += 4) {
        v2f a = *(const v2f*)(xrow + k + 2 * g);
        const int krow = k + 2 * g;
        #pragma unroll
        for (int nt = 0; nt < 4; ++nt) {
            const int col = nt * 16 + r;
            v2f b;
            b[0] = sw[(krow + 0) * Hh + col];
            b[1] = sw[(krow + 1) * Hh + col];
            acc[nt] = __builtin_amdgcn_wmma_f32_16x16x4_f32(
                false, a, false, b, (short)0, acc[nt], false, false);
        }
    }

    // D layout: VGPR j -> row M = j + 8*g, col N = nt*16 + r
    #pragma unroll
    for (int nt = 0; nt < 4; ++nt)
        #pragma unroll
        for (int j = 0; j < 8; ++j)
            hbuf[(size_t)(node0 + j + 8 * g) * VH + (size_t)v * Hh + nt * 16 + r] = acc[nt][j];
}

// ---------------- K2: degree ----------------
__global__ __launch_bounds__(256) void k_deg(const int* __restrict__ dst, int E, float* deg) {
    int e = blockIdx.x * blockDim.x + threadIdx.x;
    if (e < E) atomicAdd(&deg[dst[e]], 1.0f);
}

// ---------------- K3: dinv in place ----------------
__global__ __launch_bounds__(256) void k_dinv(float* deg) {
    int n = blockIdx.x * blockDim.x + threadIdx.x;
    if (n < Nn) { float d = deg[n]; deg[n] = (d > 0.0f) ? rsqrtf(d) : 0.0f; }
}

// ---------------- K4: edge scatter-add ----------------
__global__ __launch_bounds__(256) void k_scatter(const int* __restrict__ src,
                                                 const int* __restrict__ dst, int E,
                                                 const float* __restrict__ dinv,
                                                 const float* __restrict__ hbuf,
                                                 float* __restrict__ agg) {
    unsigned idx = blockIdx.x * 256u + threadIdx.x;
    unsigned total = (unsigned)E * (unsigned)VH;
    if (idx >= total) return;
    int e = (int)(idx / VH);
    int f = (int)(idx % VH);
    int s = src[e], d = dst[e];
    float w = dinv[s] * dinv[d];
    atomicAdd(&agg[(size_t)d * VH + f], hbuf[(size_t)s * VH + f] * w);
}

// ---------------- K5: emb = relu(agg + self-loop + bias), in place ----------------
__global__ __launch_bounds__(256) void k_emb(float* __restrict__ agg,
                                             const float* __restrict__ hbuf,
                                             const float* __restrict__ dinv,
                                             const float* __restrict__ bv) {
    size_t idx = (size_t)blockIdx.x * blockDim.x + threadIdx.x;
    if (idx >= (size_t)Nn * VH) return;
    int n = (int)(idx / VH);
    int f = (int)(idx % VH);
    float di = dinv[n];
    float val = agg[idx] + hbuf[idx] * di * di + bv[f];
    agg[idx] = fmaxf(val, 0.0f);
}

// ---------------- K6: scores + per-view max ----------------
__global__ __launch_bounds__(256) void k_scores(const float* __restrict__ emb,
                                                const float* __restrict__ na_w,
                                                const float* __restrict__ na_b,
                                                float* __restrict__ scores,
                                                unsigned* __restrict__ gmax) {
    __shared__ unsigned smax[Vv];
    int tid = threadIdx.x;
    if (tid < Vv) smax[tid] = 0u;
    __syncthreads();
    int idx = blockIdx.x * blockDim.x + tid;
    if (idx < Nn * Vv) {
        int n = idx / Vv, v = idx % Vv;
        const float* er = emb + (size_t)n * VH + (size_t)v * Hh;
        float s = na_b[0];
        for (int k = 0; k < Hh; ++k) s += er[k] * na_w[k];
        scores[idx] = s;
        atomicMax(&smax[v], enc_f(s));
    }
    __syncthreads();
    if (tid < Vv) atomicMax(&gmax[tid], smax[tid]);
}

// ---------------- K7: per-view sum of exp ----------------
__global__ __launch_bounds__(256) void k_sumexp(const float* __restrict__ scores,
                                                const unsigned* __restrict__ gmax,
                                                float* __restrict__ gsum) {
    __shared__ float ssum[Vv];
    int tid = threadIdx.x;
    if (tid < Vv) ssum[tid] = 0.0f;
    __syncthreads();
    int idx = blockIdx.x * blockDim.x + tid;
    if (idx < Nn * Vv) {
        int v = idx % Vv;
        atomicAdd(&ssum[v], expf(scores[idx] - dec_f(gmax[v])));
    }
    __syncthreads();
    if (tid < Vv) atomicAdd(&gsum[tid], ssum[tid]);
}

// ---------------- K8: avg-sum of emb_w over nodes (mean done later) ----------------
__global__ __launch_bounds__(192) void k_avg(const float* __restrict__ emb,
                                             const float* __restrict__ scores,
                                             const unsigned* __restrict__ gmax,
                                             const float* __restrict__ gsum,
                                             float* __restrict__ avg) {
    int t = threadIdx.x;          // feature index 0..191
    int v = t / Hh;
    int n0 = blockIdx.x * 64;
    float s = 0.0f;
    for (int i = 0; i < 64; ++i) {
        int n = n0 + i;
        if (n < Nn) s += emb[(size_t)n * VH + t] * attn_nv(scores, gmax, gsum, n, v);
    }
    atomicAdd(&avg[t], s);
}

// ---------------- K9: view attention MLP + softmax over views ----------------
__global__ __launch_bounds__(32) void k_view(const float* __restrict__ avg,
                                             const float* __restrict__ va_w1,
                                             const float* __restrict__ va_b1,
                                             const float* __restrict__ va_w2,
                                             const float* __restrict__ va_b2,
                                             float* __restrict__ vw,
                                             float* __restrict__ out_vw) {
    __shared__ float red[32];
    __shared__ float vs3[Vv];
    int t = threadIdx.x;          // hidden index 0..31 (H/2)
    const float invN = 1.0f / (float)Nn;
    for (int v = 0; v < Vv; ++v) {
        float a = va_b1[t];
        for (int h = 0; h < Hh; ++h) a += (avg[v * Hh + h] * invN) * va_w1[h * 32 + t];
        red[t] = tanhf(a) * va_w2[t];
        __syncthreads();
        for (int s = 16; s > 0; s >>= 1) {
            if (t < s) red[t] += red[t + s];
            __syncthreads();
        }
        if (t == 0) vs3[v] = red[0] + va_b2[0];
        __syncthreads();
    }
    if (t == 0) {
        float m = fmaxf(vs3[0], fmaxf(vs3[1], vs3[2]));
        float e0 = expf(vs3[0] - m), e1 = expf(vs3[1] - m), e2 = expf(vs3[2] - m);
        float s = e0 + e1 + e2;
        vw[0] = e0 / s; vw[1] = e1 / s; vw[2] = e2 / s;
        out_vw[0] = vw[0]; out_vw[1] = vw[1]; out_vw[2] = vw[2];
    }
}

// ---------------- K10: fused features + classifier + log_softmax ----------------
__global__ __launch_bounds__(64) void k_fused(const float* __restrict__ emb,
                                              const float* __restrict__ scores,
                                              const unsigned* __restrict__ gmax,
                                              const float* __restrict__ gsum,
                                              const float* __restrict__ vw,
                                              const float* __restrict__ cls_w,
                                              const float* __restrict__ cls_b,
                                              float* __restrict__ out) {
    __shared__ float r0[64], r1[64];
    int n = blockIdx.x;
    int t = threadIdx.x;
    float w0 = attn_nv(scores, gmax, gsum, n, 0) * vw[0];
    float w1 = attn_nv(scores, gmax, gsum, n, 1) * vw[1];
    float w2 = attn_nv(scores, gmax, gsum, n, 2) * vw[2];
    const float* er = emb + (size_t)n * VH;
    float f = er[t] * w0 + er[Hh + t] * w1 + er[2 * Hh + t] * w2;
    out[(size_t)Nn * 2 + (size_t)n * Hh + t] = f;            // fused output
    r0[t] = f * cls_w[2 * t];
    r1[t] = f * cls_w[2 * t + 1];
    __syncthreads();
    for (int s = 32; s > 0; s >>= 1) {
        if (t < s) { r0[t] += r0[t + s]; r1[t] += r1[t + s]; }
        __syncthreads();
    }
    if (t == 0) {
        float l0 = r0[0] + cls_b[0];
        float l1 = r1[0] + cls_b[1];
        float m = fmaxf(l0, l1);
        float ls = m + logf(expf(l0 - m) + expf(l1 - m));
        out[(size_t)n * 2 + 0] = l0 - ls;                    // log-softmax
        out[(size_t)n * 2 + 1] = l1 - ls;
    }
}

extern "C" void kernel_launch(void* const* d_in, const int* in_sizes, int n_in,
                              void* d_out, int out_size, void* d_ws, size_t ws_size,
                              hipStream_t stream) {
    const float* x      = (const float*)d_in[0];
    const int*   eidx   = (const int*)d_in[1];
    const float* Wv     = (const float*)d_in[2];
    const float* bv     = (const float*)d_in[3];
    const float* na_w   = (const float*)d_in[4];
    const float* na_b   = (const float*)d_in[5];
    const float* va_w1  = (const float*)d_in[6];
    const float* va_b1  = (const float*)d_in[7];
    const float* va_w2  = (const float*)d_in[8];
    const float* va_b2  = (const float*)d_in[9];
    const float* cls_w  = (const float*)d_in[10];
    const float* cls_b  = (const float*)d_in[11];
    float* out = (float*)d_out;

    const int E = in_sizes[1] / 2;
    const int* esrc = eidx;
    const int* edst = eidx + E;

    // workspace layout (floats)
    float*    ws     = (float*)d_ws;
    float*    hbuf   = ws;                               // N*V*H
    float*    agg    = hbuf + (size_t)Nn * VH;           // N*V*H (becomes emb)
    float*    dinv   = agg  + (size_t)Nn * VH;           // N (deg -> dinv)
    float*    scores = dinv + Nn;                        // N*V
    unsigned* gmax   = (unsigned*)(scores + (size_t)Nn * Vv); // V
    float*    gsum   = (float*)(gmax + Vv);              // V
    float*    avg    = gsum + Vv;                        // V*H
    float*    vw     = avg + VH;                         // V

    const size_t nvh = (size_t)Nn * VH;

    // K0 init
    {
        int blocks = (int)((nvh + 255) / 256);
        k_init<<<blocks, 256, 0, stream>>>(agg, dinv, gmax, gsum, avg);
    }
    // K1 WMMA GEMM: view-major blocks, 8 node-tiles (8 waves) per block
    {
        int blocksPerView = ((Nn / 16) + 7) / 8;         // 391
        k_gemm_wmma<<<Vv * blocksPerView, 256, 0, stream>>>(x, Wv, hbuf);
    }
    // K2 degree
    k_deg<<<(E + 255) / 256, 256, 0, stream>>>(edst, E, dinv);
    // K3 dinv
    k_dinv<<<(Nn + 255) / 256, 256, 0, stream>>>(dinv);
    // K4 scatter
    {
        unsigned total = (unsigned)E * (unsigned)VH;
        k_scatter<<<(total + 255u) / 256u, 256, 0, stream>>>(esrc, edst, E, dinv, hbuf, agg);
    }
    // K5 emb
    k_emb<<<(int)((nvh + 255) / 256), 256, 0, stream>>>(agg, hbuf, dinv, bv);
    // K6 scores + max
    k_scores<<<(Nn * Vv + 255) / 256, 256, 0, stream>>>(agg, na_w, na_b, scores, gmax);
    // K7 sumexp
    k_sumexp<<<(Nn * Vv + 255) / 256, 256, 0, stream>>>(scores, gmax, gsum);
    // K8 avg partials
    k_avg<<<(Nn + 63) / 64, 192, 0, stream>>>(agg, scores, gmax, gsum, avg);
    // K9 view attention
    k_view<<<1, 32, 0, stream>>>(avg, va_w1, va_b1, va_w2, va_b2, vw,
                                 out + (size_t)Nn * 2 + (size_t)Nn * Hh);
    // K10 fused + classifier
    k_fused<<<Nn, 64, 0, stream>>>(agg, scores, gmax, gsum, vw, cls_w, cls_b, out);
}